// DropConnect_86234353369427
// MI455X (gfx1250) — compile-verified
//
#include <hip/hip_runtime.h>

typedef __attribute__((ext_vector_type(2))) float v2f;
typedef __attribute__((ext_vector_type(8))) float v8f;

#define B_SAMPLES 128
#define IN_DIM    512
#define OUT_DIM   512

// Grid: 512 blocks = 128 samples x 4 column-groups.
// Block: 256 threads = 8 wave32s; each wave owns one 16-column output tile.
// Per wave: z[col0..col0+15] = sum_k x[k] * (W[k,col]*Werr[loc,k,col]), done as
// V_WMMA_F32_16X16X4_F32 with A = x-chunk broadcast to all 16 rows (every row
// of D computes the same partial sums; one row is read out at the end).
//
// Memory roofline: Werr gather = 128 x 1 MB = 134 MB unique HBM traffic
// (~5.8 us floor @ 23.3 TB/s); W (1 MB) and X (256 KB) are L2-resident.
// The loop is structured to maximize load issue rate: outer loop steps 32
// k-rows (one uniform prefetch, no per-iteration scalar conditions), inner
// body fully unrolled into 8 WMMAs with 32 coalesced b32 loads.
__global__ __launch_bounds__(256) void dropconnect_wmma_kernel(
    const float* __restrict__ X,
    const float* __restrict__ W,
    const float* __restrict__ bias,
    const float* __restrict__ Werr,
    const float* __restrict__ Berr,
    const int*   __restrict__ loc_id,
    float*       __restrict__ Z)
{
    __shared__ float xs[IN_DIM];

    const int sample = blockIdx.x >> 2;     // which batch row
    const int cgroup = blockIdx.x & 3;      // which 128-column group
    const int tid  = threadIdx.x;
    const int wave = tid >> 5;
    const int lane = tid & 31;

    // Stage x[sample, :] into LDS (512 floats, 2 KB): 256 threads x 2.
    const float* xrow = X + sample * IN_DIM;
    xs[tid]       = xrow[tid];
    xs[tid + 256] = xrow[tid + 256];
    __syncthreads();

    const int loc = loc_id[sample];
    const float* Eb = Werr + (size_t)loc * (size_t)(IN_DIM * OUT_DIM);

    const int col0 = (cgroup * 8 + wave) * 16;   // 16-column tile base
    const int n    = lane & 15;
    const int col  = col0 + n;
    const int koff = (lane >> 4) * 2;            // lanes 0-15: K={0,1}; 16-31: K={2,3}

    // Per-lane base pointers; k advances via immediate-foldable row offsets.
    const float* wp = W  + koff * OUT_DIM + col;
    const float* ep = Eb + koff * OUT_DIM + col;

    v8f acc = {};

    for (int k0 = 0; k0 < IN_DIM; k0 += 32) {
        // One speculative prefetch per 32-row chunk, 64 rows ahead (uniform
        // scalar address math; clamped so we never leave this Werr slab).
        int pr = k0 + 64;
        if (pr >= IN_DIM) pr = IN_DIM - 32;
        __builtin_prefetch((const void*)(Eb + pr * OUT_DIM + col), 0, 0);

        #pragma unroll
        for (int kk = 0; kk < 32; kk += 4) {
            const int r = (k0 + kk) * OUT_DIM;
            const int rx = k0 + kk + koff;

            // A fragment: x chunk broadcast across all 16 M-rows (LDS broadcast).
            v2f a;
            a.x = xs[rx];
            a.y = xs[rx + 1];

            // B fragment: masked weights (W .* Werr), rows rx,rx+1, 16 cols.
            const float w0 = wp[r];
            const float w1 = wp[r + OUT_DIM];
            const float e0 = ep[r];
            const float e1 = ep[r + OUT_DIM];
            v2f b;
            b.x = w0 * e0;
            b.y = w1 * e1;

            // D = A*B + C (16x16x4, f32). All D rows identical by construction.
            acc = __builtin_amdgcn_wmma_f32_16x16x4_f32(
                /*neg_a=*/false, a, /*neg_b=*/false, b,
                /*c_mod=*/(short)0, acc, /*reuse_a=*/false, /*reuse_b=*/false);
        }
    }

    // acc[0]: lanes 0-15 hold row M=0 (N = lane); lanes 16-31 a duplicate row.
    const float memb = bias[col] * Berr[(size_t)loc * OUT_DIM + col];
    const float z = acc[0] + memb;
    if (lane < 16) {
        Z[sample * OUT_DIM + col] = z;
    }
}

extern "C" void kernel_launch(void* const* d_in, const int* in_sizes, int n_in,
                              void* d_out, int out_size, void* d_ws, size_t ws_size,
                              hipStream_t stream) {
    const float* X    = (const float*)d_in[0];
    const float* W    = (const float*)d_in[1];
    const float* bias = (const float*)d_in[2];
    const float* Werr = (const float*)d_in[3];
    const float* Berr = (const float*)d_in[4];
    const int*   loc  = (const int*)  d_in[5];
    float* Z = (float*)d_out;

    dim3 grid(B_SAMPLES * 4);   // 512 blocks: sample * 4 column-groups
    dim3 block(256);            // 8 wave32s
    hipLaunchKernelGGL(dropconnect_wmma_kernel, grid, block, 0, stream,
                       X, W, bias, Werr, Berr, loc, Z);
}